// GaussianSceneModel_15805479649650
// MI455X (gfx1250) — compile-verified
//
#include <hip/hip_runtime.h>
#include <math.h>

// ---------------- problem constants (match setup_inputs) ----------------
#define NG      1000000
#define IMW     1600
#define IMH     900
#define LW      1024
#define LH      128
#define NEARP   1.0f
#define FARP    100.0f
#define FMIN_R  (-0.4363323129985824f)   // radians(-25)
#define FMAX_R  ( 0.05235987755982988f)  // radians(3)
#define EPSV    1e-8f
#define INV2PI  0.15915494309189535f

// d_out layout (flat, reference return order)
#define R_RGB   0
#define R_DEP   (IMH*IMW*3)              // 4,320,000
#define R_ALP   (R_DEP + IMH*IMW)        // 5,760,000
#define R_LD    (R_ALP + IMH*IMW)        // 7,200,000
#define R_LA    (R_LD + LH*LW)           // 7,331,072
#define TOT_OUT (R_LA + LH*LW)           // 7,462,144
#define PTOT    (IMH*IMW + LH*LW)        // finalize threads

typedef __attribute__((ext_vector_type(2))) float v2f;
typedef __attribute__((ext_vector_type(8))) float v8f;

// ---------------- 4x4 inverse (Gauss-Jordan, partial pivot) -------------
__device__ inline void inv4(const float* M, float out[16]) {
    float a[4][8];
    for (int i = 0; i < 4; ++i)
        for (int j = 0; j < 4; ++j) { a[i][j] = M[i*4+j]; a[i][4+j] = (i == j) ? 1.f : 0.f; }
    for (int c = 0; c < 4; ++c) {
        int piv = c; float best = fabsf(a[c][c]);
        for (int r = c+1; r < 4; ++r) { float v = fabsf(a[r][c]); if (v > best) { best = v; piv = r; } }
        if (piv != c) for (int j = 0; j < 8; ++j) { float t = a[c][j]; a[c][j] = a[piv][j]; a[piv][j] = t; }
        float inv = 1.f / a[c][c];
        for (int j = 0; j < 8; ++j) a[c][j] *= inv;
        for (int r = 0; r < 4; ++r) if (r != c) {
            float f = a[r][c];
            for (int j = 0; j < 8; ++j) a[r][j] -= f * a[c][j];
        }
    }
    for (int i = 0; i < 4; ++i) for (int j = 0; j < 4; ++j) out[i*4+j] = a[i][4+j];
}

// Build A-matrix (16x4, row-major) for the WMMA: rows 0-2 = w2c[0:3,:],
// rows 3-5 = w2l[0:3,:], rest zero.
__global__ void prep_kernel(const float* __restrict__ c2w,
                            const float* __restrict__ l2w,
                            float* __restrict__ Amat) {
    float w2c[16], w2l[16];
    inv4(c2w, w2c);
    inv4(l2w, w2l);
    for (int i = 0; i < 64; ++i) Amat[i] = 0.f;
    for (int r = 0; r < 3; ++r)
        for (int k = 0; k < 4; ++k) {
            Amat[r*4 + k]     = w2c[r*4 + k];
            Amat[(r+3)*4 + k] = w2l[r*4 + k];
        }
}

__global__ void zero_kernel(float* __restrict__ out, int n) {
    int i = blockIdx.x * blockDim.x + threadIdx.x;
    if (i < n) out[i] = 0.f;
}

__device__ inline float sigmoidf_(float x) { return 1.f / (1.f + expf(-x)); }

// ---------------- WMMA transform + atomic bilinear scatter --------------
__global__ __launch_bounds__(256)
void scatter_kernel(const float* __restrict__ means,
                    const float* __restrict__ sh,
                    const float* __restrict__ opaC,
                    const float* __restrict__ opaL,
                    const float* __restrict__ Kmat,
                    const float* __restrict__ Amat,
                    float* __restrict__ out) {
    const int tid  = blockIdx.x * 256 + threadIdx.x;
    const int lane = threadIdx.x & 31;
    const int wbase = tid - lane;          // first gaussian of this wave
    const int j = lane & 15;
    const bool hi = lane >= 16;

    // A (16x4 f32): lane m holds A[m,0],A[m,1]; lane m+16 holds A[m,2],A[m,3]
    v2f A;
    {
        int kh = hi ? 2 : 0;
        A.x = Amat[j*4 + kh];
        A.y = Amat[j*4 + kh + 1];
    }
    // B (4x16 f32): lane n holds B[0,n],B[1,n] = (x,y); lane n+16 holds (z,1)
    const int p1 = wbase + j;
    const int p2 = p1 + 16;
    v2f B1, B2;
    if (!hi) {
        B1.x = (p1 < NG) ? means[3*p1]     : 0.f;
        B1.y = (p1 < NG) ? means[3*p1 + 1] : 0.f;
        B2.x = (p2 < NG) ? means[3*p2]     : 0.f;
        B2.y = (p2 < NG) ? means[3*p2 + 1] : 0.f;
    } else {
        B1.x = (p1 < NG) ? means[3*p1 + 2] : 0.f;
        B1.y = 1.f;
        B2.x = (p2 < NG) ? means[3*p2 + 2] : 0.f;
        B2.y = 1.f;
    }
    v8f C = {};
    // D rows 0-2 = camera-space xyz, rows 3-5 = lidar-space xyz, cols = points
    v8f D1 = __builtin_amdgcn_wmma_f32_16x16x4_f32(false, A, false, B1, (short)0, C, false, false);
    v8f D2 = __builtin_amdgcn_wmma_f32_16x16x4_f32(false, A, false, B2, (short)0, C, false, false);

    // lane L<16 owns gaussian p1 (from D1 col j); lane L>=16 owns p2 whose
    // D2 columns live in lanes 0-15 -> pull them over with shuffles.
    float vtx[6];
#pragma unroll
    for (int r = 0; r < 6; ++r) {
        float s = __shfl(D2[r], j, 32);
        vtx[r] = hi ? s : D1[r];
    }
    const float cxp = vtx[0], cyp = vtx[1], czp = vtx[2];
    const float lxp = vtx[3], lyp = vtx[4], lzp = vtx[5];

    const int g = tid;
    if (g >= NG) return;

    // ---------------- camera splat ----------------
    {
        const bool validc = (czp > NEARP) && (czp < FARP);
        const float zs = validc ? czp : 1.f;
        const float fx = Kmat[0], cx0 = Kmat[2], fy = Kmat[4], cy0 = Kmat[5];
        const float u = fx * cxp / zs + cx0;
        const float v = fy * cyp / zs + cy0;
        float oc = fminf(fmaxf(opaC[g], -20.f), 20.f);
        const float wc = sigmoidf_(oc) * (validc ? 1.f : 0.f);
        if (wc > 0.f) {
            const float r0 = sigmoidf_(sh[g*48 + 0]);
            const float g0 = sigmoidf_(sh[g*48 + 1]);
            const float b0 = sigmoidf_(sh[g*48 + 2]);
            const float u0 = floorf(u), v0 = floorf(v);
            const float fu = u - u0, fv = v - v0;
            const int u0i = (int)u0, v0i = (int)v0;
            const float cw[4] = {(1.f-fu)*(1.f-fv), fu*(1.f-fv), (1.f-fu)*fv, fu*fv};
            const int du[4] = {0,1,0,1}, dv[4] = {0,0,1,1};
#pragma unroll
            for (int c = 0; c < 4; ++c) {
                const int ui = u0i + du[c];
                const int vi = v0i + dv[c];
                if (ui >= 0 && ui < IMW && vi >= 0 && vi < IMH) {
                    const float ww = wc * cw[c];
                    if (ww != 0.f) {
                        const int idx = vi * IMW + ui;
                        atomicAdd(&out[R_RGB + idx*3 + 0], r0 * ww);
                        atomicAdd(&out[R_RGB + idx*3 + 1], g0 * ww);
                        atomicAdd(&out[R_RGB + idx*3 + 2], b0 * ww);
                        atomicAdd(&out[R_DEP + idx], czp * ww);
                        atomicAdd(&out[R_ALP + idx], ww);
                    }
                }
            }
        }
    }

    // ---------------- lidar splat ----------------
    {
        const float rr = sqrtf(lxp*lxp + lyp*lyp + lzp*lzp);
        const float az = atan2f(lyp, lxp);
        float ratio = lzp / fmaxf(rr, 1e-6f);
        ratio = fminf(fmaxf(ratio, -1.f), 1.f);
        const float el = asinf(ratio);
        const float uL = (az * INV2PI + 0.5f) * (float)LW;
        const float vL = (FMAX_R - el) / (FMAX_R - FMIN_R) * (float)(LH - 1);
        const bool validl = (rr > NEARP) && (rr < FARP) && (el >= FMIN_R) && (el <= FMAX_R);
        float ol = fminf(fmaxf(opaL[g], -20.f), 20.f);
        const float wl = sigmoidf_(ol) * (validl ? 1.f : 0.f);
        if (wl > 0.f) {
            const float u0 = floorf(uL), v0 = floorf(vL);
            const float fu = uL - u0, fv = vL - v0;
            const int u0i = (int)u0, v0i = (int)v0;
            const float cw[4] = {(1.f-fu)*(1.f-fv), fu*(1.f-fv), (1.f-fu)*fv, fu*fv};
            const int du[4] = {0,1,0,1}, dv[4] = {0,0,1,1};
#pragma unroll
            for (int c = 0; c < 4; ++c) {
                const int ui = (u0i + du[c]) & (LW - 1);   // wrap (LW is pow2)
                const int vi = v0i + dv[c];
                if (vi >= 0 && vi < LH) {
                    const float ww = wl * cw[c];
                    if (ww != 0.f) {
                        const int idx = vi * LW + ui;
                        atomicAdd(&out[R_LD + idx], rr * ww);
                        atomicAdd(&out[R_LA + idx], ww);
                    }
                }
            }
        }
    }
}

// ---------------- in-place normalization ----------------
__global__ void finalize_kernel(float* __restrict__ out) {
    const int p = blockIdx.x * blockDim.x + threadIdx.x;
    if (p < IMH*IMW) {
        const float w = out[R_ALP + p];
        const float a = fminf(w, 1.f);                 // w >= 0 always
        const float inv = 1.f / (w + EPSV);
        out[R_RGB + p*3 + 0] *= inv * a;
        out[R_RGB + p*3 + 1] *= inv * a;
        out[R_RGB + p*3 + 2] *= inv * a;
        out[R_DEP + p] *= inv;
        out[R_ALP + p] = a;
    } else if (p < PTOT) {
        const int q = p - IMH*IMW;
        const float w = out[R_LA + q];
        out[R_LD + q] = out[R_LD + q] / (w + EPSV);
        out[R_LA + q] = fminf(w, 1.f);
    }
}

extern "C" void kernel_launch(void* const* d_in, const int* in_sizes, int n_in,
                              void* d_out, int out_size, void* d_ws, size_t ws_size,
                              hipStream_t stream) {
    const float* means = (const float*)d_in[0];
    // d_in[1] log_scales, d_in[2] quats: unused by reference forward
    const float* sh    = (const float*)d_in[3];
    const float* opaC  = (const float*)d_in[4];
    const float* opaL  = (const float*)d_in[5];
    const float* c2w   = (const float*)d_in[6];
    const float* Kmat  = (const float*)d_in[7];
    const float* l2w   = (const float*)d_in[8];
    float* out  = (float*)d_out;
    float* Amat = (float*)d_ws;   // 64 floats

    prep_kernel<<<1, 1, 0, stream>>>(c2w, l2w, Amat);
    zero_kernel<<<(TOT_OUT + 255)/256, 256, 0, stream>>>(out, TOT_OUT);
    scatter_kernel<<<(NG + 255)/256, 256, 0, stream>>>(means, sh, opaC, opaL, Kmat, Amat, out);
    finalize_kernel<<<(PTOT + 255)/256, 256, 0, stream>>>(out);
}